// MSROIPooler_33466385170742
// MI455X (gfx1250) — compile-verified
//
#include <hip/hip_runtime.h>
#include <stdint.h>

// MSROIPooler for MI455X (gfx1250).
// Roofline: ~418MB of output @ 23.3TB/s => ~20us floor; all feature maps
// (~45MB) are L2-resident (192MB L2). So: stream NT stores, keep gathers in
// cache, hoist all coordinate math out of the 256-channel loop.
// CDNA5 paths: per-block TDM (tensor_load_to_lds + s_wait_tensorcnt) stages
// the block's box row into LDS; global_prefetch_b8 warms the feature plane.
// Gathers are paired: clamped 2x2 bilinear footprint => 2x global_load_b64
// per sample instead of 4x b32 (halves vector-memory request pressure).

typedef uint32_t u32;
typedef __attribute__((ext_vector_type(4))) uint32_t u32x4;
typedef __attribute__((ext_vector_type(8))) uint32_t u32x8;

static constexpr int kC = 256;  // channels (fixed by reference)

template <int P, int HSZ, int BINS>
__global__ __launch_bounds__(256) void roi_align_level(
    const float* __restrict__ feat,   // [N, C, HSZ, HSZ]
    const float* __restrict__ boxes,  // [M, 5] = (batch, x1, y1, x2, y2)
    float* __restrict__ out,          // [M, C, P, P]
    float scale) {
  constexpr int HW = HSZ * HSZ;
  constexpr int SAMPLES = BINS * 4;       // 2x2 samples per bin
  constexpr int CHUNKS = (P * P) / BINS;  // bin-chunks per box

  __shared__ float sbox[8];  // this block's box row (5 floats, padded)
  __shared__ int s_off[2][SAMPLES];   // top-pair / bottom-pair element offsets
  __shared__ float s_w[4][SAMPLES];   // w00,w01,w10,w11 (incl. 1/4 sample mean)

  const int tid = threadIdx.x;
  const int box = blockIdx.x / CHUNKS;
  const int chunk = blockIdx.x % CHUNKS;

  // ---- Phase 0: TDM async copy of this box's row (5 f32 = 20B) into LDS ----
  if (tid < 32) {  // one wave issues; TDM ignores EXEC; TENSORcnt is per-wave
    const u32 lds_addr = (u32)(size_t)(void*)sbox;  // low 32 bits = LDS offset
    const uint64_t ga = (uint64_t)(size_t)(boxes + (size_t)box * 5);

    // D# group 0 (128b): count=1 | lds_addr | global_addr[56:0] | type=2
    u32x4 g0;
    g0[0] = 1u;
    g0[1] = lds_addr;
    g0[2] = (u32)(ga & 0xFFFFFFFFull);
    g0[3] = ((u32)(ga >> 32) & 0x01FFFFFFu) | (2u << 30);

    // D# group 1 (256b): data_size=4B, tensor 5x1, tile 5x1, stride0=5
    u32x8 g1;
    g1[0] = (2u << 16);   // wg_mask=0, data_size=2 (4 bytes)
    g1[1] = (5u << 16);   // atomic_barrier_addr=0 | tensor_dim0[15:0]=5
    g1[2] = (1u << 16);   // tensor_dim0[31:16]=0 | tensor_dim1[15:0]=1
    g1[3] = (5u << 16);   // tensor_dim1[31:16]=0 | tile_dim0=5
    g1[4] = 1u;           // tile_dim1=1 | tile_dim2=0
    g1[5] = 5u;           // tensor_dim0_stride[31:0]=5
    g1[6] = 0u;           // stride0[47:32]=0 | tensor_dim1_stride[15:0]=0
    g1[7] = 0u;           // tensor_dim1_stride[47:16]=0

    asm volatile("tensor_load_to_lds %0, %1" ::"s"(g0), "s"(g1) : "memory");
    __builtin_amdgcn_s_wait_tensorcnt(0);
  }
  __syncthreads();

  const int b = (int)sbox[0];
  const float rx1 = sbox[1] * scale;
  const float ry1 = sbox[2] * scale;
  const float rx2 = sbox[3] * scale;
  const float ry2 = sbox[4] * scale;
  const float roi_w = fmaxf(rx2 - rx1, 1.0f);
  const float roi_h = fmaxf(ry2 - ry1, 1.0f);
  const float bin_w = roi_w / (float)P;
  const float bin_h = roi_h / (float)P;

  const float* fbase = feat + (size_t)b * kC * HW;
  if (tid == 0) __builtin_prefetch(fbase, 0, 1);  // global_prefetch_b8

  // ---- Phase 1: per-(bin,sample) bilinear taps -> LDS (shared by 256 ch) ----
  for (int i = tid; i < SAMPLES; i += 256) {
    const int lbin = i >> 2;
    const int s = i & 3;
    const int gbin = chunk * BINS + lbin;
    const int py = gbin / P, px = gbin % P;
    const float qy = ((float)(s >> 1) + 0.5f) * 0.5f;  // (q+0.5)/2, q in {0,1}
    const float qx = ((float)(s & 1) + 0.5f) * 0.5f;
    const float yy = ry1 + ((float)py + qy) * bin_h;
    const float xx = rx1 + ((float)px + qx) * bin_w;
    const bool valid =
        (yy >= -1.0f) && (yy <= (float)HSZ) && (xx >= -1.0f) && (xx <= (float)HSZ);
    const float y = fminf(fmaxf(yy, 0.0f), (float)(HSZ - 1));
    const float x = fminf(fmaxf(xx, 0.0f), (float)(HSZ - 1));
    // Clamp the 2x2 footprint base to [0, HSZ-2] and fold the edge case into
    // the fraction: identical bilinear result, but taps are always an
    // in-bounds 2x2 block with two adjacent-row pairs (b64-loadable).
    const int y0 = min((int)floorf(y), HSZ - 2);
    const int x0 = min((int)floorf(x), HSZ - 2);
    const float ly = y - (float)y0, lx = x - (float)x0;  // in [0,1]
    const float hy = 1.0f - ly, hx = 1.0f - lx;
    const float vm = valid ? 0.25f : 0.0f;  // folds the 2x2-sample mean
    s_off[0][i] = y0 * HSZ + x0;        // top pair (x0, x0+1)
    s_off[1][i] = (y0 + 1) * HSZ + x0;  // bottom pair
    s_w[0][i] = hy * hx * vm;
    s_w[1][i] = hy * lx * vm;
    s_w[2][i] = ly * hx * vm;
    s_w[3][i] = ly * lx * vm;
  }
  __syncthreads();

  // ---- Phase 2: channel streaming. thread -> fixed bin, strided channels ----
  const int bin = tid % BINS;  // bin-fastest => coalesced NT stores
  const int c0 = tid / BINS;
  constexpr int CSTEP = 256 / BINS;
  const int gbin = chunk * BINS + bin;

  int ot[4], ob[4];
  float w[4][4];
#pragma unroll
  for (int s = 0; s < 4; ++s) {
    const int i = bin * 4 + s;
    ot[s] = s_off[0][i];
    ob[s] = s_off[1][i];
#pragma unroll
    for (int t = 0; t < 4; ++t) w[s][t] = s_w[t][i];
  }

  float* obase = out + (size_t)box * kC * (P * P) + gbin;
#pragma unroll 2
  for (int c = c0; c < kC; c += CSTEP) {
    const float* p = fbase + (size_t)c * HW;
    float acc = 0.0f;
#pragma unroll
    for (int s = 0; s < 4; ++s) {
      const float2 t = *(const float2*)(p + ot[s]);  // global_load_b64
      const float2 bo = *(const float2*)(p + ob[s]); // global_load_b64
      acc = fmaf(w[s][0], t.x, acc);
      acc = fmaf(w[s][1], t.y, acc);
      acc = fmaf(w[s][2], bo.x, acc);
      acc = fmaf(w[s][3], bo.y, acc);
    }
    __builtin_nontemporal_store(acc, obase + (size_t)c * (P * P));  // stream out
  }
}

extern "C" void kernel_launch(void* const* d_in, const int* in_sizes, int n_in,
                              void* d_out, int out_size, void* d_ws, size_t ws_size,
                              hipStream_t stream) {
  (void)n_in; (void)out_size; (void)d_ws; (void)ws_size;
  const float* x0 = (const float*)d_in[0];  // [2,256,128,128]
  const float* x1 = (const float*)d_in[1];  // [2,256,64,64]
  const float* x2 = (const float*)d_in[2];  // [2,256,32,32]
  const float* x3 = (const float*)d_in[3];  // [2,256,16,16]
  const float* boxes = (const float*)d_in[4];
  const int M = in_sizes[4] / 5;
  float* out = (float*)d_out;

  const size_t n0 = (size_t)M * kC * 4 * 4;
  const size_t n1 = (size_t)M * kC * 8 * 8;
  const size_t n2 = (size_t)M * kC * 16 * 16;

  dim3 blk(256);
  // <P, H(==W), BINS-per-block>; chunks/box = P*P/BINS
  roi_align_level<4, 128, 16><<<dim3(M * 1), blk, 0, stream>>>(x0, boxes, out, 0.125f);
  roi_align_level<8, 64, 64><<<dim3(M * 1), blk, 0, stream>>>(x1, boxes, out + n0, 0.0625f);
  roi_align_level<16, 32, 64><<<dim3(M * 4), blk, 0, stream>>>(x2, boxes, out + n0 + n1, 0.03125f);
  roi_align_level<32, 16, 64><<<dim3(M * 16), blk, 0, stream>>>(x3, boxes, out + n0 + n1 + n2, 0.015625f);
}